// CAM_8615704396004
// MI455X (gfx1250) — compile-verified
//
#include <hip/hip_runtime.h>

// ---------------------------------------------------------------------------
// Channel-attention (SE/CBAM) block for MI455X (gfx1250, wave32).
//
// Roofline: x = 32*512*56*56 fp32 = 205.5MB. Min HBM traffic ~= 2 reads +
// 1 write of x ~= 616MB -> ~26us @ 23.3TB/s. We cut the second read by
// walking the scale pass in reverse macro-order (x nearly fits the 192MB L2)
// and keep the write stream out of L2 with non-temporal stores.
// The 512->32->512 MLP (~4 MFLOP) runs on V_WMMA_F32_16X16X4_F32 in fp32.
// ---------------------------------------------------------------------------

typedef __attribute__((ext_vector_type(2))) float v2f;
typedef __attribute__((ext_vector_type(4))) float v4f;
typedef __attribute__((ext_vector_type(8))) float v8f;

#define C_IN    512
#define HID     32
#define BATCH   32
#define HW      3136                  // 56*56
#define NPLANE  (BATCH * C_IN)        // 16384 (b,c) planes
#define NVEC4   ((size_t)NPLANE * (HW / 4))   // 12,845,056 float4s
#define V4_PER_PLANE (HW / 4)         // 784

static __device__ __forceinline__ v8f wmma_f32_k4(v2f a, v2f b, v8f acc) {
  // D = A(16x4 f32) x B(4x16 f32) + C(16x16 f32)
  return __builtin_amdgcn_wmma_f32_16x16x4_f32(
      /*neg_a=*/false, a, /*neg_b=*/false, b,
      /*c_mod=*/(short)0, acc, /*reuse_a=*/false, /*reuse_b=*/false);
}

// ---------------------------------------------------------------------------
// Kernel 1: per-plane avg + max pooling.  P[64][512]: rows 0-31 avg, 32-63 max.
// One 256-thread block per (b,c) plane; 784 float4 loads, wave32 shfl reduce.
// ---------------------------------------------------------------------------
__global__ __launch_bounds__(256) void cam_pool_kernel(
    const float* __restrict__ x, float* __restrict__ P) {
  const int plane = blockIdx.x;                // b*512 + c
  const v4f* px4 = (const v4f*)(x + (size_t)plane * HW);
  const int tid = threadIdx.x;

  float s = 0.0f;
  float m = -__builtin_huge_valf();
#pragma unroll
  for (int i = 0; i < 3; ++i) {                // 3*256 = 768 float4s
    v4f v = px4[tid + 256 * i];
    s += (v.x + v.y) + (v.z + v.w);
    m = fmaxf(m, fmaxf(fmaxf(v.x, v.y), fmaxf(v.z, v.w)));
  }
  if (tid < 16) {                              // tail: 784 - 768
    v4f v = px4[768 + tid];
    s += (v.x + v.y) + (v.z + v.w);
    m = fmaxf(m, fmaxf(fmaxf(v.x, v.y), fmaxf(v.z, v.w)));
  }
  // wave32 reduction
#pragma unroll
  for (int off = 16; off > 0; off >>= 1) {
    s += __shfl_xor(s, off, 32);
    m = fmaxf(m, __shfl_xor(m, off, 32));
  }
  __shared__ float ss[8], sm[8];
  if ((tid & 31) == 0) { ss[tid >> 5] = s; sm[tid >> 5] = m; }
  __syncthreads();
  if (tid == 0) {
    float S = ss[0], M = sm[0];
#pragma unroll
    for (int w = 1; w < 8; ++w) { S += ss[w]; M = fmaxf(M, sm[w]); }
    const int b = plane >> 9;                  // /512
    const int c = plane & 511;
    P[b * C_IN + c]            = S * (1.0f / (float)HW);  // avg
    P[(b + BATCH) * C_IN + c]  = M;                        // max
  }
}

// ---------------------------------------------------------------------------
// Kernel 2: gate = sigmoid(mlp(avg) + mlp(max)) entirely on WMMA f32 16x16x4.
// Layer 1:  H(64x32) = relu( P(64x512) @ w1^T(512x32) )       [w1 is 32x512]
// Layer 2:  G(32x512) = [H_avg | H_max](32x64) @ [w2^T;w2^T](64x512)
// Single block, 8 waves. Fragment layouts per ISA 7.12.2:
//   A 16x4: lane<16 -> M=lane, {v0,v1}={K0,K1}; lane>=16 -> M=lane-16,{K2,K3}
//   B 4x16: lane<16 -> N=lane, {v0,v1}={K0,K1}; lane>=16 -> N=lane-16,{K2,K3}
//   C 16x16: lane<16 -> N=lane, vgpr r -> M=r;  lane>=16 -> N=lane-16, M=8+r
// ---------------------------------------------------------------------------
__global__ __launch_bounds__(256) void cam_mlp_gate_kernel(
    const float* __restrict__ P, const float* __restrict__ w1,
    const float* __restrict__ w2, float* __restrict__ gate) {
  __shared__ float Hs[64 * HID];               // 8 KB
  const int tid  = threadIdx.x;
  const int wave = tid >> 5;
  const int lane = tid & 31;
  const int half = lane >> 4;                  // selects K-pair / +8 rows
  const int idx  = lane & 15;

  // ----- Layer 1: 4 M-tiles x 2 N-tiles = 8 tiles -> one per wave ----------
  {
    const int ti = wave >> 1;                  // M tile: rows 16*ti..
    const int tj = wave & 1;                   // N tile over hidden 32
    v8f acc = {};
    const float* arow = P  + (size_t)(16 * ti + idx) * C_IN + 2 * half;
    const float* brow = w1 + (size_t)(16 * tj + idx) * C_IN + 2 * half;
    for (int k0 = 0; k0 < C_IN; k0 += 4) {
      v2f a = *(const v2f*)(arow + k0);        // A[m][k0+2h .. +1]
      v2f b = *(const v2f*)(brow + k0);        // B[k][n] = w1[n][k]
      acc = wmma_f32_k4(a, b, acc);
    }
#pragma unroll
    for (int r = 0; r < 8; ++r) {              // ReLU -> LDS H[64][32]
      const int row = 16 * ti + 8 * half + r;
      Hs[row * HID + 16 * tj + idx] = fmaxf(acc[r], 0.0f);
    }
  }
  __syncthreads();

  // ----- Layer 2: 2 M-tiles x 32 N-tiles = 64 tiles -> 8 per wave ----------
  for (int t = wave; t < 64; t += 8) {
    const int bi = t & 1;                      // batch tile (rows 16*bi..)
    const int cj = t >> 1;                     // channel tile (cols 16*cj..)
    v8f acc = {};
#pragma unroll
    for (int k0 = 0; k0 < 64; k0 += 4) {
      const int ksub = k0 & 31;                // K stays within a 32-block
      const int hoff = (k0 < 32) ? 0 : 32;     // avg rows vs max rows of H
      v2f a = *(const v2f*)(&Hs[(16 * bi + idx + hoff) * HID + ksub + 2 * half]);
      v2f b = *(const v2f*)(w2 + (size_t)(16 * cj + idx) * HID + ksub + 2 * half);
      acc = wmma_f32_k4(a, b, acc);
    }
#pragma unroll
    for (int r = 0; r < 8; ++r) {
      const int brow = 16 * bi + 8 * half + r;
      const float g = 1.0f / (1.0f + __expf(-acc[r]));
      gate[brow * C_IN + 16 * cj + idx] = g;
    }
  }
}

// ---------------------------------------------------------------------------
// Kernel 3: out = x * gate[plane].  One float4 per thread.  Blocks run in
// REVERSE order vs kernel 1's read order so the L2-resident tail of x hits;
// output stored non-temporally so the write stream doesn't evict x.
// ---------------------------------------------------------------------------
__global__ __launch_bounds__(256) void cam_scale_kernel(
    const float* __restrict__ x, const float* __restrict__ gate,
    float* __restrict__ out) {
  const unsigned rb = gridDim.x - 1u - blockIdx.x;       // reversed macro order
  const size_t i = (size_t)rb * 256 + threadIdx.x;       // float4 index
  const unsigned plane = (unsigned)(i / V4_PER_PLANE);   // = b*512 + c
  const float g = gate[plane];                           // L1/L2 cached scalar
  v4f v = ((const v4f*)x)[i];
  v4f o;
  o.x = v.x * g; o.y = v.y * g; o.z = v.z * g; o.w = v.w * g;
  __builtin_nontemporal_store(o, (v4f*)out + i);
}

// ---------------------------------------------------------------------------
extern "C" void kernel_launch(void* const* d_in, const int* in_sizes, int n_in,
                              void* d_out, int out_size, void* d_ws, size_t ws_size,
                              hipStream_t stream) {
  const float* x  = (const float*)d_in[0];   // [32,512,56,56]
  const float* w1 = (const float*)d_in[1];   // [32,512]
  const float* w2 = (const float*)d_in[2];   // [512,32]
  float* out = (float*)d_out;

  float* P    = (float*)d_ws;                // [64][512]  avg rows 0-31, max 32-63
  float* gate = P + 64 * C_IN;               // [32][512]

  cam_pool_kernel<<<NPLANE, 256, 0, stream>>>(x, P);
  cam_mlp_gate_kernel<<<1, 256, 0, stream>>>(P, w1, w2, gate);
  cam_scale_kernel<<<(unsigned)(NVEC4 / 256), 256, 0, stream>>>(x, gate, out);
}